// TensorProductCuda_51376398795039
// MI455X (gfx1250) — compile-verified
//
#include <hip/hip_runtime.h>
#include <stdint.h>

typedef __attribute__((ext_vector_type(2))) float v2f;
typedef __attribute__((ext_vector_type(8))) float v8f;
typedef __attribute__((ext_vector_type(4))) int   v4i;

#define DIM_IN 16
#define NCH 16
#define WAVES 8
#define XS_STRIDE 17   // pad to dodge LDS bank conflicts on x_s[m][c] gathers
#define MAX_TILES 32

// out[e] (dim3 x 16) = A (dim3 x nnz, fixed one-hot by mu3) @ B(e) (nnz x 16)
// A = one-hot indicator (segment scatter done by the matrix unit),
// B(e)[k][c] = cg[k] * W[c,widx[k]] * y[e,mu2[k]] * x[e,mu1[k],c].
// K padded per 16-row m3 tile to a multiple of 8 -> branch-free, 2x-unrolled
// inner loop (two V_WMMA_F32_16X16X4_F32 per iteration).

__device__ __forceinline__ void wmma_step(v8f& acc, v4i r, int m3tgt,
                                          const float* __restrict__ Wli,
                                          const float* __restrict__ yw,
                                          const float* __restrict__ xw,
                                          int li) {
  v2f av, bv;
  {
    uint32_t meta = (uint32_t)r.x; float c = __uint_as_float(r.y);
    int m1 = meta & 255, m2 = (meta >> 8) & 255, m3v = (meta >> 16) & 255, p = meta >> 24;
    bv[0] = c * Wli[p * NCH] * yw[m2] * xw[m1 * XS_STRIDE + li];
    av[0] = (m3v == m3tgt) ? 1.0f : 0.0f;
  }
  {
    uint32_t meta = (uint32_t)r.z; float c = __uint_as_float(r.w);
    int m1 = meta & 255, m2 = (meta >> 8) & 255, m3v = (meta >> 16) & 255, p = meta >> 24;
    bv[1] = c * Wli[p * NCH] * yw[m2] * xw[m1 * XS_STRIDE + li];
    av[1] = (m3v == m3tgt) ? 1.0f : 0.0f;
  }
  acc = __builtin_amdgcn_wmma_f32_16x16x4_f32(false, av, false, bv, (short)0,
                                              acc, false, false);
}

__device__ __forceinline__ v8f tile_accum(const uint32_t* __restrict__ rec,
                                          int ks, int ke, int m3tgt, int hf,
                                          const float* __restrict__ Wli,
                                          const float* __restrict__ yw,
                                          const float* __restrict__ xw,
                                          int li) {
  v8f acc = {0.f, 0.f, 0.f, 0.f, 0.f, 0.f, 0.f, 0.f};
  for (int k0 = ks; k0 < ke; k0 += 8) {
    // two b128 LDS loads: each covers this lane's two K-slots (16B aligned)
    v4i r0 = *(const v4i*)(rec + (unsigned)(k0 + 2 * hf) * 2u);
    v4i r1 = *(const v4i*)(rec + (unsigned)(k0 + 4 + 2 * hf) * 2u);
    wmma_step(acc, r0, m3tgt, Wli, yw, xw, li);
    wmma_step(acc, r1, m3tgt, Wli, yw, xw, li);
  }
  return acc;
}

__global__ __launch_bounds__(256) void tp_wmma_kernel(
    const float* __restrict__ x,   // (E, 16, 16)
    const float* __restrict__ y,   // (E, 16)
    const float* __restrict__ cg,  // (nnz)
    const float* __restrict__ W,   // (16, n_paths)
    const int*   __restrict__ mu1,
    const int*   __restrict__ mu2,
    const int*   __restrict__ mu3, // globally non-decreasing
    const int*   __restrict__ widx,
    float*       __restrict__ out, // (E, dim3, 16)
    int E, int nnz, int n_paths, int dim3_, int ntiles, int rec_cap)
{
  extern __shared__ char smem[];
  uint32_t* rec = (uint32_t*)smem;                         // rec_cap records * 2 u32
  float*    Wt  = (float*)(smem + (size_t)rec_cap * 8);    // n_paths x 16 (transposed)
  float*    x_s = Wt + (size_t)n_paths * NCH;              // WAVES * 16*17
  float*    y_s = x_s + WAVES * DIM_IN * XS_STRIDE;        // WAVES * 16
  __shared__ int kb_s[MAX_TILES + 1];
  __shared__ int pkb[MAX_TILES + 1];

  const int tid  = threadIdx.x;
  const int wave = tid >> 5;
  const int lane = tid & 31;
  const int hf   = lane >> 4;   // wave half (K split for A/B frags)
  const int li   = lane & 15;   // channel / row-within-half index

  // ---- per-block setup -------------------------------------------------
  for (int t = tid; t <= ntiles; t += blockDim.x) {
    int target = t * 16, lo = 0, hi = nnz;
    while (lo < hi) { int mid = (lo + hi) >> 1; if (mu3[mid] < target) lo = mid + 1; else hi = mid; }
    kb_s[t] = lo;
  }
  __syncthreads();
  if (tid == 0) {            // padded (multiple-of-8) prefix offsets
    int acc = 0; pkb[0] = 0;
    for (int t = 0; t < ntiles; ++t) {
      int sz = kb_s[t + 1] - kb_s[t];
      acc += (sz + 7) & ~7;
      pkb[t + 1] = acc;
    }
  }
  __syncthreads();
  for (int t = 0; t < ntiles; ++t) {   // packed+padded records
    const int ks = kb_s[t], sz = kb_s[t + 1] - ks;
    const int base = pkb[t], psz = pkb[t + 1] - base;
    for (int i = tid; i < psz; i += blockDim.x) {
      uint32_t meta; float c;
      if (i < sz) {
        int k = ks + i;
        meta = (uint32_t)mu1[k] | ((uint32_t)mu2[k] << 8) |
               ((uint32_t)mu3[k] << 16) | ((uint32_t)widx[k] << 24);
        c = cg[k];
      } else {
        meta = 0x00FF0000u;  // mu3=255 never matches; cg=0 kills B too
        c = 0.0f;
      }
      rec[(unsigned)(base + i) * 2u]      = meta;
      rec[(unsigned)(base + i) * 2u + 1u] = __float_as_uint(c);
    }
  }
  for (int i = tid; i < n_paths * NCH; i += blockDim.x) {  // Wt[p][c] = W[c][p]
    int p = i >> 4, c = i & 15;
    Wt[p * NCH + c] = W[c * n_paths + p];
  }
  __syncthreads();

  // ---- per-wave edge loop (all offsets fit 32 bits: sizes are int) -----
  float* xw = x_s + wave * (DIM_IN * XS_STRIDE);
  float* yw = y_s + wave * DIM_IN;
  const float* Wli = Wt + li;
  const int stride = (int)gridDim.x * WAVES;
  const int full   = dim3_ >> 4;          // tiles with all 16 rows valid
  const int tailrows = dim3_ & 15;

  for (int e = (int)blockIdx.x * WAVES + wave; e < E; e += stride) {
    const float* xe = x + (unsigned)e * (DIM_IN * NCH);
#pragma unroll
    for (int q = 0; q < 8; ++q) {
      int m = 2 * q + hf;
      xw[m * XS_STRIDE + li] = xe[m * NCH + li];
    }
    if (lane < DIM_IN) yw[lane] = y[(unsigned)e * DIM_IN + lane];
    __builtin_amdgcn_wave_barrier();

    int en = e + stride;                  // prefetch next edge's x (8 lines)
    if (en < E && lane < 8)
      __builtin_prefetch(x + (unsigned)en * (DIM_IN * NCH) + lane * 32, 0, 1);

    // per-lane output base: row (8*hf), col li of this edge's (dim3 x 16) slab
    float* obase = out + (unsigned)e * (unsigned)(dim3_ * NCH) + (8 * hf) * NCH + li;

    for (int t = 0; t < full; ++t) {      // full tiles: 8 unguarded NT stores
      v8f acc = tile_accum(rec, pkb[t], pkb[t + 1], t * 16 + li, hf, Wli, yw, xw, li);
      float* optr = obase + t * (16 * NCH);
#pragma unroll
      for (int i = 0; i < 8; ++i)
        __builtin_nontemporal_store(acc[i], optr + i * NCH);
    }
    if (tailrows) {                       // single partial tile: guarded rows
      int t = full;
      v8f acc = tile_accum(rec, pkb[t], pkb[t + 1], t * 16 + li, hf, Wli, yw, xw, li);
      float* optr = obase + t * (16 * NCH);
#pragma unroll
      for (int i = 0; i < 8; ++i)
        if (8 * hf + i < tailrows)
          __builtin_nontemporal_store(acc[i], optr + i * NCH);
    }
    __builtin_amdgcn_wave_barrier();
  }
}

extern "C" void kernel_launch(void* const* d_in, const int* in_sizes, int n_in,
                              void* d_out, int out_size, void* d_ws, size_t ws_size,
                              hipStream_t stream) {
  const float* x    = (const float*)d_in[0];
  const float* y    = (const float*)d_in[1];
  const float* cg   = (const float*)d_in[2];
  const float* W    = (const float*)d_in[3];
  const int*   mu1  = (const int*)d_in[4];
  const int*   mu2  = (const int*)d_in[5];
  const int*   mu3  = (const int*)d_in[6];
  const int*   widx = (const int*)d_in[7];
  // d_in[8] is the dim3 scalar on device; derive everything host-side.

  const int E       = in_sizes[0] / (DIM_IN * NCH);
  const int nnz     = in_sizes[2];
  const int n_paths = in_sizes[3] / NCH;
  const int dim3_   = out_size / (E * NCH);
  int ntiles        = (dim3_ + 15) / 16;
  if (ntiles > MAX_TILES) ntiles = MAX_TILES;

  const int rec_cap = nnz + 8 * ntiles;  // upper bound incl. per-tile padding
  const size_t shmem = (size_t)rec_cap * 8                    // packed records
                     + (size_t)n_paths * NCH * 4              // transposed W
                     + (size_t)WAVES * DIM_IN * XS_STRIDE * 4 // x_s
                     + (size_t)WAVES * DIM_IN * 4;            // y_s

  const int blocks = (E + WAVES - 1) / WAVES;  // 1 edge per wave (grid-stride safe)
  tp_wmma_kernel<<<blocks, 256, shmem, stream>>>(
      x, y, cg, W, mu1, mu2, mu3, widx, (float*)d_out,
      E, nnz, n_paths, dim3_, ntiles, rec_cap);
}